// DotProductAttention_25958782337182
// MI455X (gfx1250) — compile-verified
//
#include <hip/hip_runtime.h>
#include <hip/hip_bf16.h>

#define B_   32
#define T_   2048
#define D_   1024
#define TC   128            // t-rows per workgroup chunk
#define NCH  (T_ / TC)      // 16 chunks per batch

typedef __attribute__((ext_vector_type(2))) float v2f;
typedef __attribute__((ext_vector_type(8))) float v8f;

// ---------------------------------------------------------------------------
// Kernel 1: per (batch b, chunk c of 128 t-rows):
//   - scores s_t = q . V[b,t,:]  via V_WMMA_F32_16X16X4_F32 (16 rows / wave)
//   - chunk softmax partials m_c, Z_c ; raw scores -> weights output region
//   - partial context acc_c[d] = sum_t exp(s_t - m_c) * V[b,t,d]  (L2-hot re-read)
// ---------------------------------------------------------------------------
__global__ __launch_bounds__(256) void attn_chunk_kernel(
    const float* __restrict__ q,      // [B, D]
    const float* __restrict__ v,      // [B, T, D]
    float* __restrict__ out_w_raw,    // [B, T]   (raw scores for now)
    float* __restrict__ ws_acc,       // [B, NCH, D]
    float* __restrict__ ws_m,         // [B, NCH]
    float* __restrict__ ws_z)         // [B, NCH]
{
    __shared__ __align__(16) float q_s[D_];
    __shared__ float sc_s[TC];
    __shared__ float p_s[TC];
    __shared__ float red_s[256];

    const int c    = blockIdx.x;       // chunk
    const int b    = blockIdx.y;       // batch
    const int tid  = threadIdx.x;
    const int lane = tid & 31;         // wave32
    const int wave = tid >> 5;         // 0..7
    const int t0   = c * TC;

    // ---- stage query into LDS (256 threads x float4 = 1024 floats) ----
    {
        const float4* qg = (const float4*)(q + (size_t)b * D_);
        ((float4*)q_s)[tid] = qg[tid];
    }
    __syncthreads();

    // ---- WMMA score matvec: wave handles rows t0 + wave*16 .. +15 ----
    // A (16x4 f32) layout: lanes 0-15 -> M=lane, K=0(V0),1(V1); lanes 16-31 -> M=lane-16, K=2,3
    // B (4x16 f32): q broadcast across all 16 columns -> every D column holds the scores
    const int   row  = t0 + wave * 16 + (lane & 15);
    const int   koff = (lane >> 4) << 1;                 // 0 or 2
    const float* vrow = v + ((size_t)b * T_ + row) * D_ + koff;
    const float* qk   = q_s + koff;

    v8f acc = {};
#pragma unroll 8
    for (int k = 0; k < D_; k += 4) {
        v2f a  = *(const v2f*)(vrow + k);   // global, 8B per lane
        v2f bq = *(const v2f*)(qk + k);     // LDS broadcast
        acc = __builtin_amdgcn_wmma_f32_16x16x4_f32(
            /*neg_a=*/false, a, /*neg_b=*/false, bq,
            /*c_mod=*/(short)0, acc, /*reuse_a=*/false, /*reuse_b=*/false);
    }
    // D layout: lane<16 has M=r in acc[r] (any column n identical); lane>=16 has M=8+r
    if (lane == 0) {
#pragma unroll
        for (int r = 0; r < 8; ++r) sc_s[wave * 16 + r] = acc[r];
    } else if (lane == 16) {
#pragma unroll
        for (int r = 0; r < 8; ++r) sc_s[wave * 16 + 8 + r] = acc[r];
    }
    __syncthreads();

    // ---- chunk softmax partials (deterministic tree reductions) ----
    float sv = (tid < TC) ? sc_s[tid] : -3.402823466e38f;
    if (tid < TC) out_w_raw[(size_t)b * T_ + t0 + tid] = sv;   // stash raw score
    red_s[tid] = sv;
    __syncthreads();
#pragma unroll
    for (int off = 128; off > 0; off >>= 1) {
        if (tid < off) red_s[tid] = fmaxf(red_s[tid], red_s[tid + off]);
        __syncthreads();
    }
    const float mc = red_s[0];
    __syncthreads();                        // everyone read mc before reuse

    float pv = 0.0f;
    if (tid < TC) { pv = expf(sv - mc); p_s[tid] = pv; }
    red_s[tid] = pv;
    __syncthreads();
#pragma unroll
    for (int off = 128; off > 0; off >>= 1) {
        if (tid < off) red_s[tid] += red_s[tid + off];
        __syncthreads();
    }
    if (tid == 0) {
        ws_m[b * NCH + c] = mc;
        ws_z[b * NCH + c] = red_s[0];
    }

    // ---- partial context: re-read this 512KB chunk (L2/L0-hot), coalesced ----
    // thread tid owns d = 4*tid .. 4*tid+3
    float4 a4 = make_float4(0.f, 0.f, 0.f, 0.f);
    const float4* vb = (const float4*)(v + ((size_t)b * T_ + t0) * D_);
    for (int i = 0; i < TC; ++i) {
        const float  pw = p_s[i];
        const float4 x  = vb[(size_t)i * (D_ / 4) + tid];
        a4.x += pw * x.x; a4.y += pw * x.y; a4.z += pw * x.z; a4.w += pw * x.w;
    }
    ((float4*)(ws_acc + (size_t)(b * NCH + c) * D_))[tid] = a4;
}

// ---------------------------------------------------------------------------
// Kernel 2: per batch b, merge the NCH chunk partials (log-sum-exp combine),
// write context, normalize the weights in place.
// ---------------------------------------------------------------------------
__global__ __launch_bounds__(256) void attn_combine_kernel(
    const float* __restrict__ ws_acc,   // [B, NCH, D]
    const float* __restrict__ ws_m,     // [B, NCH]
    const float* __restrict__ ws_z,     // [B, NCH]
    float* __restrict__ out_ctx,        // [B, D]
    float* __restrict__ out_w)          // [B, T] raw in, normalized out
{
    const int b   = blockIdx.x;
    const int tid = threadIdx.x;

    // every thread redundantly computes the (tiny) combine factors
    float mloc[NCH];
    float M = -3.402823466e38f;
#pragma unroll
    for (int c = 0; c < NCH; ++c) {
        mloc[c] = ws_m[b * NCH + c];
        M = fmaxf(M, mloc[c]);
    }
    float f[NCH];
    float Z = 0.0f;
#pragma unroll
    for (int c = 0; c < NCH; ++c) {
        f[c] = expf(mloc[c] - M);
        Z += ws_z[b * NCH + c] * f[c];
    }
    const float invZ = 1.0f / Z;

    // context: thread tid owns d = 4*tid .. 4*tid+3
    float4 s4 = make_float4(0.f, 0.f, 0.f, 0.f);
#pragma unroll
    for (int c = 0; c < NCH; ++c) {
        const float4 x = ((const float4*)(ws_acc + (size_t)(b * NCH + c) * D_))[tid];
        s4.x += f[c] * x.x; s4.y += f[c] * x.y; s4.z += f[c] * x.z; s4.w += f[c] * x.w;
    }
    s4.x *= invZ; s4.y *= invZ; s4.z *= invZ; s4.w *= invZ;
    ((float4*)(out_ctx + (size_t)b * D_))[tid] = s4;

    // normalize attention weights (each t touched by exactly one thread)
    for (int t = tid; t < T_; t += 256) {
        const float s = out_w[(size_t)b * T_ + t];
        out_w[(size_t)b * T_ + t] = expf(s - M) * invZ;
    }
}

extern "C" void kernel_launch(void* const* d_in, const int* in_sizes, int n_in,
                              void* d_out, int out_size, void* d_ws, size_t ws_size,
                              hipStream_t stream) {
    (void)in_sizes; (void)n_in; (void)out_size; (void)ws_size;
    const float* q = (const float*)d_in[0];     // [B, D]
    const float* v = (const float*)d_in[1];     // [B, T, D]

    float* out_ctx = (float*)d_out;                          // [B, D]
    float* out_w   = (float*)d_out + (size_t)B_ * D_;        // [B, T]

    float* ws_acc = (float*)d_ws;                            // B*NCH*D floats (2 MB)
    float* ws_m   = ws_acc + (size_t)B_ * NCH * D_;          // B*NCH
    float* ws_z   = ws_m + B_ * NCH;                         // B*NCH

    dim3 g1(NCH, B_);
    attn_chunk_kernel<<<g1, 256, 0, stream>>>(q, v, out_w, ws_acc, ws_m, ws_z);
    attn_combine_kernel<<<dim3(B_), 256, 0, stream>>>(ws_acc, ws_m, ws_z, out_ctx, out_w);
}